// GCN_layer_52407190945898
// MI455X (gfx1250) — compile-verified
//
#include <hip/hip_runtime.h>

typedef float v2f __attribute__((ext_vector_type(2)));
typedef float v8f __attribute__((ext_vector_type(8)));

#define N_DIM 64

// ---------------------------------------------------------------------------
// Kernel 1: init — zero accumulator (d_out), deg = 1.0 (self-loop folded in)
// ---------------------------------------------------------------------------
__global__ void gcn_init_kernel(float* __restrict__ deg, float* __restrict__ acc,
                                int n_nodes) {
    int i = blockIdx.x * blockDim.x + threadIdx.x;
    if (i < n_nodes * N_DIM) acc[i] = 0.0f;
    if (i < n_nodes)         deg[i] = 1.0f;
}

// ---------------------------------------------------------------------------
// Kernel 2: degree of target nodes (f32 atomic adds of 1.0 are exact/order-free)
// ---------------------------------------------------------------------------
__global__ void gcn_degree_kernel(const int* __restrict__ col,
                                  float* __restrict__ deg, int n_edges) {
    int e = blockIdx.x * blockDim.x + threadIdx.x;
    if (e < n_edges) atomicAdd(&deg[col[e]], 1.0f);
}

// ---------------------------------------------------------------------------
// Kernel 3: xw = x @ W via V_WMMA_F32_16X16X4_F32.
// One wave computes a 16x64 output tile. W is staged in LDS in a
// pair-interleaved layout so each B operand {W[k][c], W[k+1][c]} is one
// aligned ds_load_b64 straight into an even VGPR pair (no repack movs).
// ---------------------------------------------------------------------------
__global__ __launch_bounds__(256) void gcn_gemm_wmma_kernel(
    const float* __restrict__ x, const float* __restrict__ W,
    float* __restrict__ xw, int n_nodes) {
    // ldsW2[p*64 + c] = { W[2p][c], W[2p+1][c] },  p = 0..31, c = 0..63
    __shared__ v2f ldsW2[(N_DIM / 2) * N_DIM];    // 16 KB
    for (int i = threadIdx.x; i < (N_DIM / 2) * N_DIM; i += blockDim.x) {
        const int p = i >> 6;
        const int c = i & 63;
        v2f w;
        w.x = W[(2 * p) * N_DIM + c];
        w.y = W[(2 * p + 1) * N_DIM + c];
        ldsW2[i] = w;
    }
    __syncthreads();

    const int lane = threadIdx.x & 31;
    const int wave = threadIdx.x >> 5;
    const int tile = blockIdx.x * 8 + wave;       // 16-row tile index
    if (tile * 16 >= n_nodes) return;             // wave-uniform: EXEC stays all-1

    const int m0   = tile * 16;
    const int half = lane >> 4;                   // 0 -> K pair {0,1}, 1 -> {2,3}
    const int idx  = lane & 15;

    const float* xr = x + (size_t)(m0 + idx) * N_DIM;

    v8f acc[4] = {v8f{}, v8f{}, v8f{}, v8f{}};
#pragma unroll
    for (int k0 = 0; k0 < N_DIM; k0 += 4) {
        const int ka = k0 + 2 * half;             // even: base K of this half's pair
        const int kp = ka >> 1;                   // pair index into ldsW2
        v2f a;                                    // A: 16x4, row m0+idx
        a.x = xr[ka + 0];
        a.y = xr[ka + 1];
#pragma unroll
        for (int nt = 0; nt < 4; ++nt) {          // 4 column tiles, reuse A
            const v2f b = ldsW2[kp * N_DIM + nt * 16 + idx];   // one ds_load_b64
            acc[nt] = __builtin_amdgcn_wmma_f32_16x16x4_f32(
                false, a, false, b, (short)0, acc[nt], false, false);
        }
    }

#pragma unroll
    for (int nt = 0; nt < 4; ++nt) {
#pragma unroll
        for (int p = 0; p < 8; ++p) {             // D: row p + 8*half, col idx
            const int row = m0 + p + 8 * half;
            xw[(size_t)row * N_DIM + nt * 16 + idx] = acc[nt][p];
        }
    }
}

// ---------------------------------------------------------------------------
// Kernel 4: per-edge gather + scale + scatter-add. 16 lanes per edge:
// one float4 gather + 4x global_atomic_add_f32 per lane.
// ---------------------------------------------------------------------------
__global__ void gcn_edge_scatter_kernel(const int* __restrict__ ei,
                                        const float* __restrict__ deg,
                                        const float* __restrict__ xw,
                                        float* __restrict__ acc, int n_edges) {
    long long t = (long long)blockIdx.x * blockDim.x + threadIdx.x;
    int e = (int)(t >> 4);
    if (e >= n_edges) return;
    int part = ((int)t & 15) * 4;

    int r = ei[e];                // source
    int c = ei[n_edges + e];      // target
    float norm = rsqrtf(deg[r]) * rsqrtf(deg[c]);

    const float4 v = *(const float4*)(xw + (size_t)r * N_DIM + part);
    float* dst = acc + (size_t)c * N_DIM + part;
    atomicAdd(dst + 0, norm * v.x);
    atomicAdd(dst + 1, norm * v.y);
    atomicAdd(dst + 2, norm * v.z);
    atomicAdd(dst + 3, norm * v.w);
}

// ---------------------------------------------------------------------------
// Kernel 5: out = relu(acc + xw/deg (self-loop msg) + b)
// ---------------------------------------------------------------------------
__global__ void gcn_finalize_kernel(const float* __restrict__ xw,
                                    const float* __restrict__ deg,
                                    const float* __restrict__ b,
                                    float* __restrict__ out, int n_nodes) {
    int i = blockIdx.x * blockDim.x + threadIdx.x;
    if (i >= n_nodes * N_DIM) return;
    int node = i >> 6;
    int d    = i & 63;
    float v = out[i] + xw[i] / deg[node] + b[d];
    out[i] = v > 0.0f ? v : 0.0f;
}

// ---------------------------------------------------------------------------
extern "C" void kernel_launch(void* const* d_in, const int* in_sizes, int n_in,
                              void* d_out, int out_size, void* d_ws, size_t ws_size,
                              hipStream_t stream) {
    const float* x  = (const float*)d_in[0];   // [N, 64]
    const int*   ei = (const int*)d_in[1];     // [2, E]
    const float* W  = (const float*)d_in[2];   // [64, 64]
    const float* b  = (const float*)d_in[3];   // [64]
    float* out = (float*)d_out;                // [N, 64] (also scatter accumulator)

    const int n_nodes = in_sizes[0] / N_DIM;
    const int n_edges = in_sizes[1] / 2;

    // workspace: deg [N] floats, then xw [N*64] floats
    float* deg = (float*)d_ws;
    size_t deg_bytes = (((size_t)n_nodes * sizeof(float)) + 255) & ~(size_t)255;
    float* xw = (float*)((char*)d_ws + deg_bytes);

    const int total = n_nodes * N_DIM;

    gcn_init_kernel<<<(total + 255) / 256, 256, 0, stream>>>(deg, out, n_nodes);

    gcn_degree_kernel<<<(n_edges + 255) / 256, 256, 0, stream>>>(
        ei + n_edges, deg, n_edges);

    const int tiles = (n_nodes + 15) / 16;
    gcn_gemm_wmma_kernel<<<(tiles + 7) / 8, 256, 0, stream>>>(x, W, xw, n_nodes);

    long long sthreads = (long long)n_edges * 16;
    gcn_edge_scatter_kernel<<<(int)((sthreads + 255) / 256), 256, 0, stream>>>(
        ei, deg, xw, out, n_edges);

    gcn_finalize_kernel<<<(total + 255) / 256, 256, 0, stream>>>(
        xw, deg, b, out, n_nodes);
}